// StandardConv_1099511628115
// MI455X (gfx1250) — compile-verified
//
#include <hip/hip_runtime.h>

#define D_SRC 128
#define D_IN  256
#define D_OUT 256
#define LDS_STRIDE 260   // 16-row h tile, padded so column reads hit distinct LDS banks

typedef __attribute__((ext_vector_type(2))) float v2f;
typedef __attribute__((ext_vector_type(8))) float v8f;

// ---------------- Kernel 1: zero workspace (agg + count) ----------------
__global__ void init_ws_kernel(float* __restrict__ ws, long long n) {
    long long i = (long long)blockIdx.x * blockDim.x + threadIdx.x;
    long long stride = (long long)gridDim.x * blockDim.x;
    for (; i < n; i += stride) ws[i] = 0.0f;
}

// ---------------- Kernel 2: edge gather + scatter-add ----------------
// 32 lanes per edge; each lane handles a float4 chunk of the 128-wide feature.
__global__ void scatter_edges_kernel(const float* __restrict__ x_src,
                                     const int* __restrict__ edge_index,
                                     float* __restrict__ agg,
                                     float* __restrict__ cnt,
                                     int E) {
    long long gtid = (long long)blockIdx.x * blockDim.x + threadIdx.x;
    int e    = (int)(gtid >> 5);
    int lane = (int)(gtid & 31);
    if (e >= E) return;
    int src = edge_index[e];
    int dst = edge_index[E + e];
    const float4 v = *(const float4*)(x_src + (long long)src * D_SRC + lane * 4);
    float* dp = agg + (long long)dst * D_SRC + lane * 4;
    atomicAdd(dp + 0, v.x);
    atomicAdd(dp + 1, v.y);
    atomicAdd(dp + 2, v.z);
    atomicAdd(dp + 3, v.w);
    if (lane == 0) atomicAdd(cnt + dst, 1.0f);
}

// ---------------- Kernel 3: fused mean + concat + GEMM(WMMA f32) + bias + ReLU ----------------
// Block = 16 nodes x 256 outputs. 8 waves; each wave owns two 16x16 output tiles.
__global__ void __launch_bounds__(256)
fused_gemm_kernel(const float* __restrict__ x_dst,
                  const float* __restrict__ agg,
                  const float* __restrict__ cnt,
                  const float* __restrict__ W,     // [D_OUT][D_IN] row-major
                  const float* __restrict__ bias,  // [D_OUT]
                  float* __restrict__ out,         // [N][D_OUT]
                  int N) {
    __shared__ float h[16 * LDS_STRIDE];

    const int t = threadIdx.x;
    const int node0 = blockIdx.x * 16;
    const bool full_tile = (node0 + 16 <= N);    // block-uniform

    // Phase 1: build h[r][0:128]=x_dst, h[r][128:256]=agg/max(count,1)
    for (int idx = t; idx < 16 * (D_IN / 4); idx += 256) {
        int r = idx >> 6;           // 0..15
        int c = (idx & 63) * 4;     // 0..252
        int node = node0 + r;
        float4 v = make_float4(0.0f, 0.0f, 0.0f, 0.0f);
        if (node < N) {
            if (c < D_SRC) {
                v = *(const float4*)(x_dst + (long long)node * D_SRC + c);
            } else {
                int k = c - D_SRC;
                float4 s = *(const float4*)(agg + (long long)node * D_SRC + k);
                float invc = 1.0f / fmaxf(cnt[node], 1.0f);
                v = make_float4(s.x * invc, s.y * invc, s.z * invc, s.w * invc);
            }
        }
        *(float4*)(h + r * LDS_STRIDE + c) = v;
    }
    __syncthreads();

    const int wave = t >> 5;       // 0..7 (wave32)
    const int lane = t & 31;
    const int M  = lane & 15;      // A-row / B-col / D-col index within tile
    const int hi = lane >> 4;      // 0: K=k0,k0+1  1: K=k0+2,k0+3 (per 16x4 f32 layout)

    for (int nt = wave; nt < D_OUT / 16; nt += 8) {
        const int n0  = nt * 16;
        const int col = n0 + M;
        const float* wrow = W + (long long)col * D_IN;   // B[k][col] = W[col][k]

        v8f acc = {};
        #pragma unroll 8
        for (int k0 = 0; k0 < D_IN; k0 += 4) {
            const int kb = k0 + hi * 2;
            v2f a = *(const v2f*)(h + M * LDS_STRIDE + kb);   // A 16x4 tile (LDS)
            v2f b = *(const v2f*)(wrow + kb);                 // B 4x16 tile (global/L2)
            acc = __builtin_amdgcn_wmma_f32_16x16x4_f32(
                /*neg_a=*/false, a, /*neg_b=*/false, b,
                /*c_mod=*/(short)0, acc, /*reuse_a=*/false, /*reuse_b=*/false);
        }

        const float bv = bias[col];
        float* obase = out + (long long)(node0 + hi * 8) * D_OUT + col;
        if (full_tile) {
            // All 16 rows in range (always true for N % 16 == 0): straight stores.
            #pragma unroll
            for (int v = 0; v < 8; ++v) {
                obase[(long long)v * D_OUT] = fmaxf(acc[v] + bv, 0.0f);
            }
        } else {
            #pragma unroll
            for (int v = 0; v < 8; ++v) {
                int node = node0 + v + hi * 8;
                if (node < N) {
                    out[(long long)node * D_OUT + col] = fmaxf(acc[v] + bv, 0.0f);
                }
            }
        }
    }
}

// ---------------- Launch ----------------
extern "C" void kernel_launch(void* const* d_in, const int* in_sizes, int n_in,
                              void* d_out, int out_size, void* d_ws, size_t ws_size,
                              hipStream_t stream) {
    const float* x_src      = (const float*)d_in[0];
    const float* x_dst      = (const float*)d_in[1];
    const int*   edge_index = (const int*)d_in[2];
    const float* W          = (const float*)d_in[3];
    const float* b          = (const float*)d_in[4];
    float* out = (float*)d_out;

    const int N = in_sizes[1] / D_SRC;   // 100000
    const int E = in_sizes[2] / 2;       // 600000

    float* agg = (float*)d_ws;
    float* cnt = agg + (long long)N * D_SRC;
    const long long ws_elems = (long long)N * D_SRC + N;

    init_ws_kernel<<<2048, 256, 0, stream>>>(agg, ws_elems);

    const long long sthreads = (long long)E * 32;
    const int sblocks = (int)((sthreads + 255) / 256);
    scatter_edges_kernel<<<sblocks, 256, 0, stream>>>(x_src, edge_index, agg, cnt, E);

    const int gblocks = (N + 15) / 16;   // 6250
    fused_gemm_kernel<<<gblocks, 256, 0, stream>>>(x_dst, agg, cnt, W, b, out, N);
}